// GATLayerImp4_10599979287266
// MI455X (gfx1250) — compile-verified
//
#include <hip/hip_runtime.h>
#include <hip/hip_bf16.h>

#define D_IN   128
#define HEADS  8
#define F_OUT  64
#define SC     16   // score columns: 0..7 = source scores, 8..15 = target scores
#define LDS_PITCH 130  // 16 rows of 128 padded to avoid 64-bank conflicts

typedef __attribute__((ext_vector_type(2))) float v2f;
typedef __attribute__((ext_vector_type(8))) float v8f;

// ---------------------------------------------------------------- utilities
__device__ __forceinline__ void atomicMaxF32(float* addr, float v) {
  // Monotone int-order trick: positives compare as signed ints,
  // negatives compare reversed as unsigned ints. Init value is -inf.
  if (v >= 0.0f) atomicMax((int*)addr, __float_as_int(v));
  else           atomicMin((unsigned int*)addr, __float_as_uint(v));
}

// ------------------------------------------------- 1) reset segment buffers
__global__ void init_segments(float* __restrict__ segmax,
                              float* __restrict__ segsum, int n) {
  int i = blockIdx.x * blockDim.x + threadIdx.x;
  if (i < n) {
    segmax[i] = __int_as_float(0xff800000);  // -inf
    segsum[i] = 0.0f;
  }
}

// ------------------------- 2) fold scoring vectors into projection columns
// wcombT[n][d], n in [0,16): n<8 -> src head n ; n>=8 -> trg head n-8
__global__ void fold_scoring(const float* __restrict__ W,      // [128, 512]
                             const float* __restrict__ s_src,  // [8, 64]
                             const float* __restrict__ s_trg,  // [8, 64]
                             float* __restrict__ wcombT) {     // [16, 128]
  int i = blockIdx.x * blockDim.x + threadIdx.x;  // 0..2047
  if (i >= SC * D_IN) return;
  int n = i >> 7, d = i & 127;
  int h = n & 7;
  const float* s = (n < 8 ? s_src : s_trg) + h * F_OUT;
  const float* w = W + (size_t)d * (HEADS * F_OUT) + h * F_OUT;
  float acc = 0.0f;
#pragma unroll 8
  for (int f = 0; f < F_OUT; ++f) acc = fmaf(w[f], s[f], acc);
  wcombT[n * D_IN + d] = acc;
}

// ------------------------------------- 3) skinny GEMM: scores = X @ wcomb
// One wave per 16-node tile; C[16x16] accumulated by 32x V_WMMA_F32_16X16X4_F32.
__global__ __launch_bounds__(256) void score_gemm(
    const float* __restrict__ X,       // [N, 128]
    const float* __restrict__ wcombT,  // [16, 128]  (B transposed: row n = col n)
    float* __restrict__ scores,        // [N, 16]
    int ntiles) {
  __shared__ float lb[SC * LDS_PITCH];
  for (int i = threadIdx.x; i < SC * D_IN; i += blockDim.x)
    lb[(i >> 7) * LDS_PITCH + (i & 127)] = wcombT[i];
  __syncthreads();

  int wave = threadIdx.x >> 5;
  int lane = threadIdx.x & 31;
  int tile = blockIdx.x * (blockDim.x >> 5) + wave;
  if (tile >= ntiles) return;  // wave-uniform: EXEC stays all-ones for WMMA

  int mn   = lane & 15;         // A-row (M) and B-col (N) owned by this lane
  int koff = (lane >> 4) << 1;  // lanes 16..31 hold K+2,K+3 of each K-quad

  const float* xrow = X + ((size_t)tile * 16 + mn) * D_IN;
  const float* brow = &lb[mn * LDS_PITCH];

  v8f c = {};
#pragma unroll
  for (int kb = 0; kb < D_IN; kb += 4) {
    v2f a = *(const v2f*)(xrow + kb + koff);   // A[16x4] slice (global, b64)
    v2f b = *(const v2f*)(brow + kb + koff);   // B[4x16] slice (LDS, ds_load_b64)
    c = __builtin_amdgcn_wmma_f32_16x16x4_f32(false, a, false, b,
                                              (short)0, c, false, false);
  }

  // C layout: VGPR r -> row (r + 8*(lane>>4)), col (lane&15)
  int rbase = tile * 16 + ((lane >> 4) << 3);
#pragma unroll
  for (int r = 0; r < 8; ++r)
    scores[(size_t)(rbase + r) * SC + mn] = c[r];
}

// ------------------- 4) edge scores + LeakyReLU + segment max (per target)
__global__ void edge_score_max(const int* __restrict__ ei, int E,
                               const float* __restrict__ scores,  // [N,16]
                               float* __restrict__ out,           // [E,8]
                               float* __restrict__ segmax) {      // [N,8]
  int e = blockIdx.x * blockDim.x + threadIdx.x;
  if (e >= E) return;
  int s = ei[e], t = ei[E + e];
  const float4* ps = (const float4*)(scores + (size_t)s * SC);       // src cols 0..7
  const float4* pt = (const float4*)(scores + (size_t)t * SC + 8);   // trg cols 8..15
  float4 a0 = ps[0], a1 = ps[1];
  float4 b0 = pt[0], b1 = pt[1];
  float v[8] = {a0.x + b0.x, a0.y + b0.y, a0.z + b0.z, a0.w + b0.w,
                a1.x + b1.x, a1.y + b1.y, a1.z + b1.z, a1.w + b1.w};
  float* sm = segmax + (size_t)t * 8;
  float* po = out + (size_t)e * 8;
#pragma unroll
  for (int h = 0; h < 8; ++h) {
    float x = v[h];
    x = x > 0.0f ? x : 0.2f * x;  // LeakyReLU(0.2)
    po[h] = x;
    atomicMaxF32(&sm[h], x);
  }
}

// --------------------------- 5) exp(score - segmax) + segment sum (atomic)
__global__ void edge_exp_sum(const int* __restrict__ ei, int E,
                             const float* __restrict__ segmax,  // [N,8]
                             float* __restrict__ out,           // [E,8]
                             float* __restrict__ segsum) {      // [N,8]
  int e = blockIdx.x * blockDim.x + threadIdx.x;
  if (e >= E) return;
  int t = ei[E + e];
  const float* mx = segmax + (size_t)t * 8;
  float* sm = segsum + (size_t)t * 8;
  float* po = out + (size_t)e * 8;
#pragma unroll
  for (int h = 0; h < 8; ++h) {
    float ex = __expf(po[h] - mx[h]);
    po[h] = ex;
    atomicAdd(&sm[h], ex);  // global_atomic_add_f32
  }
}

// --------------------------------------------------- 6) normalize in place
__global__ void edge_norm(const int* __restrict__ ei, int E,
                          const float* __restrict__ segsum,  // [N,8]
                          float* __restrict__ out) {         // [E,8]
  int e = blockIdx.x * blockDim.x + threadIdx.x;
  if (e >= E) return;
  int t = ei[E + e];
  const float* sm = segsum + (size_t)t * 8;
  float* po = out + (size_t)e * 8;
#pragma unroll
  for (int h = 0; h < 8; ++h)
    po[h] = po[h] / (sm[h] + 1e-16f);
}

// ---------------------------------------------------------------- launcher
extern "C" void kernel_launch(void* const* d_in, const int* in_sizes, int n_in,
                              void* d_out, int out_size, void* d_ws, size_t ws_size,
                              hipStream_t stream) {
  const float* X    = (const float*)d_in[0];  // node_features [N,128]
  const float* W    = (const float*)d_in[1];  // linear_proj  [128,512]
  const float* ssrc = (const float*)d_in[2];  // [1,8,64]
  const float* strg = (const float*)d_in[3];  // [1,8,64]
  const int*   ei   = (const int*)d_in[4];    // edge_index [2,E]
  float* out = (float*)d_out;                 // [E,8]

  const int N = in_sizes[0] / D_IN;
  const int E = in_sizes[4] / 2;

  // workspace layout (floats): wcombT | scores | segmax | segsum  (~12.8 MB)
  float* ws     = (float*)d_ws;
  float* wcombT = ws;                            // 16*128
  float* scores = ws + SC * D_IN;                // N*16  (offset 8 KB, 16B aligned)
  float* segmax = scores + (size_t)N * SC;       // N*8
  float* segsum = segmax + (size_t)N * HEADS;    // N*8

  const int segn = N * HEADS;
  init_segments<<<(segn + 255) / 256, 256, 0, stream>>>(segmax, segsum, segn);
  fold_scoring<<<(SC * D_IN + 255) / 256, 256, 0, stream>>>(W, ssrc, strg, wcombT);

  const int ntiles = (N + 15) / 16;          // 6250 (N divisible by 16)
  const int wpb = 256 / 32;                  // 8 waves per block
  score_gemm<<<(ntiles + wpb - 1) / wpb, 256, 0, stream>>>(X, wcombT, scores, ntiles);

  const int eb = (E + 255) / 256;
  edge_score_max<<<eb, 256, 0, stream>>>(ei, E, scores, out, segmax);
  edge_exp_sum<<<eb, 256, 0, stream>>>(ei, E, segmax, out, segsum);
  edge_norm<<<eb, 256, 0, stream>>>(ei, E, segsum, out);
}